// NormalShader_13606456394549
// MI455X (gfx1250) — compile-verified
//
#include <hip/hip_runtime.h>

// NormalShader for MI455X (gfx1250, wave32).
// HBM-bound. Mandatory streaming: p2f (33.5MB) + zbuf (33.5MB) + out (12.6MB).
// dists/bary are loaded ONLY for fragments whose softmax weight is non-zero:
// GAMMA=1e-4 makes exp((z_inv - z_max)/GAMMA) underflow to exactly 0.0f for
// all but ~1.06 of 8 fragments per pixel, so skipping their loads/gathers is
// bit-exact and cuts ~25-50MB of HBM traffic (bary is 100.7MB fully streamed).
// Floor ~7.5-8us @ 23.3 TB/s.

typedef __attribute__((ext_vector_type(4))) float v4f;
typedef __attribute__((ext_vector_type(4))) int   v4i;

namespace {
constexpr int   K         = 8;
constexpr float SIGMA_INV = 1.0e4f;                  // 1 / 1e-4
constexpr float GAMMA_INV = 1.0e4f;                  // 1 / 1e-4
constexpr float ZFAR      = 100.0f;
constexpr float ZSCALE    = 1.0f / (100.0f - 1.0f);  // 1/(ZFAR-ZNEAR)
constexpr float EPSV      = 1.0e-10f;
}

__global__ __launch_bounds__(256) void normal_shader_kernel(
    const int*   __restrict__ p2f,     // (P, K)
    const float* __restrict__ bary,    // (P, K, 3)  touched sparsely
    const float* __restrict__ zbuf,    // (P, K)
    const float* __restrict__ dists,   // (P, K)     touched sparsely
    const float* __restrict__ vnorm,   // (V, 3)     cached in WGP$/L2
    const int*   __restrict__ faces,   // (F, 3)     cached in WGP$/L2
    float*       __restrict__ out,     // (P, 3)
    int P)
{
  const int stride = gridDim.x * blockDim.x;
  for (int p = blockIdx.x * blockDim.x + threadIdx.x; p < P; p += stride) {
    // Prefetch next iteration's mandatory streams (global_prefetch_b8).
    const int pn = p + stride;
    if (pn < P) {
      __builtin_prefetch(p2f  + pn * K, 0, 0);
      __builtin_prefetch(zbuf + pn * K, 0, 0);
    }

    const int base  = p * K;       // 32B-aligned chunk per pixel
    const int base3 = base * 3;

    // Mandatory streaming loads (non-temporal: read-once).
    v4i fa = __builtin_nontemporal_load((const v4i*)(p2f  + base));
    v4i fb = __builtin_nontemporal_load((const v4i*)(p2f  + base + 4));
    v4f za = __builtin_nontemporal_load((const v4f*)(zbuf + base));
    v4f zb = __builtin_nontemporal_load((const v4f*)(zbuf + base + 4));

    const int   fid[K] = {fa.x, fa.y, fa.z, fa.w, fb.x, fb.y, fb.z, fb.w};
    const float zv [K] = {za.x, za.y, za.z, za.w, zb.x, zb.y, zb.z, zb.w};

    // Pass 1: z_inv and z_inv_max = max(max_k z_inv, EPS).
    float zinv[K];
    float zmax = EPSV;
#pragma unroll
    for (int k = 0; k < K; ++k) {
      const float zi = (fid[k] >= 0) ? (ZFAR - zv[k]) * ZSCALE : 0.0f;
      zinv[k] = zi;
      zmax    = fmaxf(zmax, zi);
    }

    // Pass 2: softmax blend. Only surviving fragments (e>0, valid) touch
    // dists/bary/faces/verts_normals.
    float wsum = 0.0f, acc0 = 0.0f, acc1 = 0.0f, acc2 = 0.0f;
#pragma unroll
    for (int k = 0; k < K; ++k) {
      const float e = expf((zinv[k] - zmax) * GAMMA_INV);  // exact 0 for most k
      if ((fid[k] >= 0) && (e > 0.0f)) {
        // prob = sigmoid(-dists/SIGMA) = 1/(1+exp(dists/SIGMA))
        const float d = __builtin_nontemporal_load(dists + base + k);
        const float w = e / (1.0f + expf(d * SIGMA_INV));
        wsum += w;

        const float b0 = __builtin_nontemporal_load(bary + base3 + 3 * k + 0);
        const float b1 = __builtin_nontemporal_load(bary + base3 + 3 * k + 1);
        const float b2 = __builtin_nontemporal_load(bary + base3 + 3 * k + 2);

        const int f  = fid[k];
        const int i0 = faces[3 * f + 0];
        const int i1 = faces[3 * f + 1];
        const int i2 = faces[3 * f + 2];
        acc0 += w * (b0 * vnorm[3 * i0 + 0] + b1 * vnorm[3 * i1 + 0] + b2 * vnorm[3 * i2 + 0]);
        acc1 += w * (b0 * vnorm[3 * i0 + 1] + b1 * vnorm[3 * i1 + 1] + b2 * vnorm[3 * i2 + 1]);
        acc2 += w * (b0 * vnorm[3 * i0 + 2] + b1 * vnorm[3 * i1 + 2] + b2 * vnorm[3 * i2 + 2]);
      }
    }

    const float delta = fmaxf(expf((EPSV - zmax) * GAMMA_INV), EPSV);
    const float rinv  = 1.0f / (wsum + delta);
    const float r0 = (acc0 + delta) * rinv;   // BG = (1,1,1)
    const float r1 = (acc1 + delta) * rinv;
    const float r2 = (acc2 + delta) * rinv;

    const float nrm = fmaxf(sqrtf(r0 * r0 + r1 * r1 + r2 * r2), 1e-12f);
    const float s   = 0.5f / nrm;             // (x/n + 1)/2 = x*(0.5/n) + 0.5

    __builtin_nontemporal_store(r0 * s + 0.5f, out + p * 3 + 0);
    __builtin_nontemporal_store(r1 * s + 0.5f, out + p * 3 + 1);
    __builtin_nontemporal_store(r2 * s + 0.5f, out + p * 3 + 2);
  }
}

extern "C" void kernel_launch(void* const* d_in, const int* in_sizes, int n_in,
                              void* d_out, int out_size, void* d_ws, size_t ws_size,
                              hipStream_t stream) {
  const int*   p2f   = (const int*)  d_in[0];   // pix_to_face (N,H,W,K)
  const float* bary  = (const float*)d_in[1];   // bary_coords (N,H,W,K,3)
  const float* zbuf  = (const float*)d_in[2];   // zbuf        (N,H,W,K)
  const float* dists = (const float*)d_in[3];   // dists       (N,H,W,K)
  const float* vnorm = (const float*)d_in[4];   // verts_normals (V,3)
  const int*   faces = (const int*)  d_in[5];   // faces       (F,3)
  float*       out   = (float*)d_out;           // (N,H,W,3)

  const int P = in_sizes[0] / K;                // number of pixels
  const int threads = 256;                      // 8 wave32 per block
  int blocks = (P + threads * 2 - 1) / (threads * 2);  // 2 pixels/thread
  if (blocks < 1) blocks = 1;

  normal_shader_kernel<<<blocks, threads, 0, stream>>>(
      p2f, bary, zbuf, dists, vnorm, faces, out, P);
}